// EGNN_77738908057953
// MI455X (gfx1250) — compile-verified
//
#include <hip/hip_runtime.h>
#include <math.h>

#define NP 384
#define H  128
#define NL 4

typedef float v2f __attribute__((ext_vector_type(2)));
typedef float v8f __attribute__((ext_vector_type(8)));

__device__ __forceinline__ v8f wmma_f32(v2f a, v2f b, v8f c) {
  // D(16x16,f32) = A(16x4,f32) x B(4x16,f32) + C
  return __builtin_amdgcn_wmma_f32_16x16x4_f32(false, a, false, b, (short)0, c, false, false);
}
__device__ __forceinline__ float silu_f(float v)    { return v / (1.0f + __expf(-v)); }
__device__ __forceinline__ float sigmoid_f(float v) { return 1.0f / (1.0f + __expf(-v)); }

// CDNA5 async copy: global -> LDS, 16 bytes per lane, tracked by ASYNCcnt.
__device__ __forceinline__ void async_copy_b128(unsigned lds_byte, const float* gaddr) {
  asm volatile("global_load_async_to_lds_b128 %0, %1, off" :: "v"(lds_byte), "v"(gaddr) : "memory");
}
__device__ __forceinline__ void wait_async0() {
  asm volatile("s_wait_asynccnt 0x0" ::: "memory");
}

// ---------------- h = rowsum(emb_w) + emb_b  (49152 rows of 384) ----------------
__global__ __launch_bounds__(256) void k_init_h(const float* __restrict__ ew,
                                                const float* __restrict__ eb,
                                                float* __restrict__ h) {
  const int lane = threadIdx.x & 31;
  const int wv   = threadIdx.x >> 5;
  const int row  = blockIdx.x * 8 + wv;
  const float* p = ew + (size_t)row * NP;
  float s = 0.0f;
  for (int k = lane; k < NP; k += 32) s += p[k];
  s += __shfl_xor(s, 16, 32); s += __shfl_xor(s, 8, 32);
  s += __shfl_xor(s, 4, 32);  s += __shfl_xor(s, 2, 32);
  s += __shfl_xor(s, 1, 32);
  if (lane == 0) h[row] = s + eb[row];
}

__global__ void k_copy_x(const float* __restrict__ src, float* __restrict__ x) {
  int i = blockIdx.x * 256 + threadIdx.x;
  if (i < NP * 3) x[i] = src[i];
}

// zero xdelta, set nbuf = [h | 0], zero nan-flag
__global__ void k_prep(float* __restrict__ xd, float* __restrict__ nbuf,
                       const float* __restrict__ h, int* __restrict__ flag) {
  int idx = blockIdx.x * 256 + threadIdx.x;
  if (idx == 0) *flag = 0;
  if (idx < NP * 3) xd[idx] = 0.0f;
  if (idx < NP * 256) {
    int c = idx & 255, i = idx >> 8;
    nbuf[idx] = (c < H) ? h[i * H + c] : 0.0f;
  }
}

// ---------------- generic WMMA GEMM: out = act(A @ W[:,woff:woff+K]^T + bias) ----
// one wave per 16x16 output tile; grid = (M/16, N/16)
__global__ __launch_bounds__(32) void k_gemm(const float* __restrict__ A, int lda,
                                             const float* __restrict__ W, int ldw, int woff,
                                             const float* __restrict__ bias,
                                             float* __restrict__ out, int ldo,
                                             int K, int act) {
  const int lane = threadIdx.x & 31;
  const int r = lane & 15, hh = lane >> 4;
  const float* arow = A + (size_t)(blockIdx.x * 16 + r) * lda;
  const float* wrow = W + (size_t)(blockIdx.y * 16 + r) * ldw + woff;
  v8f acc = {};
  for (int kk = 0; kk < K; kk += 4) {
    const int kb = kk + 2 * hh;
    v2f a; a.x = arow[kb]; a.y = arow[kb + 1];
    v2f b; b.x = wrow[kb]; b.y = wrow[kb + 1];
    acc = wmma_f32(a, b, acc);
  }
  const float bv = bias ? bias[blockIdx.y * 16 + r] : 0.0f;
#pragma unroll
  for (int v = 0; v < 8; ++v) {
    float o = acc[v] + bv;
    if (act) o = silu_f(o);
    out[(size_t)(blockIdx.x * 16 + v + 8 * hh) * ldo + blockIdx.y * 16 + r] = o;
  }
}

// ---------------- fused edge kernel ------------------------------------------------
// grid (24, 8): 16 receivers i x 48 senders j per WG, chunks of 16 j.
#define TSTR 132  // LDS row stride (floats): breaks 16-way bank conflicts
// LDS float offsets
#define T1S_OFF  0
#define W2C_OFF  33792                // 128*132
#define W2E_OFF  (W2C_OFF + 16896)
#define PS_OFF   (W2E_OFF + 16896)   // 16*128
#define PES_OFF  (PS_OFF + 2048)
#define QS_OFF   (PES_OFF + 2048)
#define QES_OFF  (QS_OFF + 2048)
#define D2S_OFF  (QES_OFF + 2048)
#define INVS_OFF (D2S_OFF + 256)
#define XACC_OFF (INVS_OFF + 256)
#define SM_FLOATS (XACC_OFF + 64)     // 76352 floats = 305408 B (<320KB/WG)

__global__ __launch_bounds__(256) void k_edge(
    const float* __restrict__ x,
    const float* __restrict__ P,  const float* __restrict__ Q,
    const float* __restrict__ Pe, const float* __restrict__ Qe,
    const float* __restrict__ cw1, const float* __restrict__ cb1,
    const float* __restrict__ cw2, const float* __restrict__ cb2,
    const float* __restrict__ cw3, const float* __restrict__ cb3,
    const float* __restrict__ ew1, const float* __restrict__ eb1,
    const float* __restrict__ ew2, const float* __restrict__ eb2,
    const float* __restrict__ attw,
    float* __restrict__ xdelta, float* __restrict__ nbuf, int has_edge) {
  extern __shared__ float smem[];
  float* t1s  = smem + T1S_OFF;
  float* w2cs = smem + W2C_OFF;
  float* w2es = smem + W2E_OFF;
  float* Ps   = smem + PS_OFF;
  float* Pes  = smem + PES_OFF;
  float* Qs   = smem + QS_OFF;
  float* Qes  = smem + QES_OFF;
  float* d2s  = smem + D2S_OFF;
  float* invs = smem + INVS_OFF;
  float* xacc = smem + XACC_OFF;
  const unsigned lds0 = (unsigned)(uintptr_t)smem;

  const int tid  = threadIdx.x;
  const int lane = tid & 31, wv = tid >> 5;
  const int r = lane & 15, hh = lane >> 4;
  const int i0 = blockIdx.x * 16;
  const int jbase = blockIdx.y * 48;

  // ---- async-stage W2 matrices (padded stride) and the P tiles, once per WG ----
  for (int g = tid; g < 4096; g += 256) {                  // 128 rows x 32 chunks
    const int rr = g >> 5, c4 = (g & 31) << 2;
    async_copy_b128(lds0 + (unsigned)(W2C_OFF + rr * TSTR + c4) * 4u, cw2 + rr * H + c4);
    if (has_edge)
      async_copy_b128(lds0 + (unsigned)(W2E_OFF + rr * TSTR + c4) * 4u, ew2 + rr * H + c4);
  }
  for (int g = tid; g < 512; g += 256) {                   // 16 rows x 32 chunks
    const int rr = g >> 5, c4 = (g & 31) << 2;
    async_copy_b128(lds0 + (unsigned)(PS_OFF + rr * H + c4) * 4u, P + (size_t)(i0 + rr) * H + c4);
    if (has_edge)
      async_copy_b128(lds0 + (unsigned)(PES_OFF + rr * H + c4) * 4u, Pe + (size_t)(i0 + rr) * H + c4);
  }

  // per-thread constants for the t1-build stage (k fixed per thread)
  const int kloc = tid & 127;
  const int eb0  = tid >> 7;
  const float cw1c = cw1[kloc * 257 + 256];
  const float cb1k = cb1[kloc];
  float ew1c = 0.0f, eb1k = 0.0f;
  if (has_edge) { ew1c = ew1[kloc * 257 + 256]; eb1k = eb1[kloc]; }

  // per-lane channel constants for GEMM epilogues
  float cb2v[8], w3v[8], eb2v[8], attv[8];
#pragma unroll
  for (int nt = 0; nt < 8; ++nt) {
    const int c = nt * 16 + r;
    cb2v[nt] = cb2[c]; w3v[nt] = cw3[c];
    eb2v[nt] = has_edge ? eb2[c] : 0.0f;
    attv[nt] = has_edge ? attw[c] : 0.0f;
  }
  const float b3 = cb3[0];
  if (tid < 64) xacc[tid] = 0.0f;

  float msum[2][8];
#pragma unroll
  for (int m = 0; m < 2; ++m)
#pragma unroll
    for (int nt = 0; nt < 8; ++nt) msum[m][nt] = 0.0f;
  const int mrow[2] = { wv, wv + 8 };  // Mtile index == receiver row ii

  for (int ch = 0; ch < 3; ++ch) {
    __syncthreads();  // previous chunk fully consumed (t1s, Qs)
    const int j0 = jbase + ch * 16;
    // async-stage this chunk's Q tiles; overlap with d2 computation
    for (int g = tid; g < 512; g += 256) {
      const int rr = g >> 5, c4 = (g & 31) << 2;
      async_copy_b128(lds0 + (unsigned)(QS_OFF + rr * H + c4) * 4u, Q + (size_t)(j0 + rr) * H + c4);
      if (has_edge)
        async_copy_b128(lds0 + (unsigned)(QES_OFF + rr * H + c4) * 4u, Qe + (size_t)(j0 + rr) * H + c4);
    }
    {  // stage 1: d2 + 1/(sqrt(d2)+1), one edge per thread
      const int ii = tid >> 4, jj = tid & 15;
      const int ig = i0 + ii, jg = j0 + jj;
      const float dx = x[jg * 3 + 0] - x[ig * 3 + 0];
      const float dy = x[jg * 3 + 1] - x[ig * 3 + 1];
      const float dz = x[jg * 3 + 2] - x[ig * 3 + 2];
      const float d2 = dx * dx + dy * dy + dz * dz;
      d2s[tid] = d2;
      invs[tid] = 1.0f / (sqrtf(d2) + 1.0f);
    }
    wait_async0();
    __syncthreads();
    // stage 2: t1 = silu(P[i] + Q[j] + d2*w1c + b1)  (coord path, LDS-resident)
    for (int it = 0; it < 128; ++it) {
      const int e = eb0 + it * 2;
      const float v = Ps[(e >> 4) * H + kloc] + Qs[(e & 15) * H + kloc] + d2s[e] * cw1c + cb1k;
      t1s[e * TSTR + kloc] = silu_f(v);
    }
    __syncthreads();
    // stage 3: coord t2 GEMM (M=256 edges, N=128, K=128), all operands in LDS
    v8f acc[2][8];
#pragma unroll
    for (int m = 0; m < 2; ++m)
#pragma unroll
      for (int nt = 0; nt < 8; ++nt) acc[m][nt] = (v8f){};
    for (int kk = 0; kk < H; kk += 4) {
      const int kb = kk + 2 * hh;
      v2f a0, a1;
      { const float* p0 = &t1s[(mrow[0] * 16 + r) * TSTR + kb]; a0.x = p0[0]; a0.y = p0[1]; }
      { const float* p1 = &t1s[(mrow[1] * 16 + r) * TSTR + kb]; a1.x = p1[0]; a1.y = p1[1]; }
#pragma unroll
      for (int nt = 0; nt < 8; ++nt) {
        const float* wp = &w2cs[(nt * 16 + r) * TSTR + kb];
        v2f b; b.x = wp[0]; b.y = wp[1];
        acc[0][nt] = wmma_f32(a0, b, acc[0][nt]);
        acc[1][nt] = wmma_f32(a1, b, acc[1][nt]);
      }
    }
    // stage 4: phi = silu(t2+b2).w3 + b3 ; accumulate coordinate deltas
#pragma unroll
    for (int m = 0; m < 2; ++m) {
      float ph[8];
#pragma unroll
      for (int v = 0; v < 8; ++v) ph[v] = 0.0f;
#pragma unroll
      for (int nt = 0; nt < 8; ++nt)
#pragma unroll
        for (int v = 0; v < 8; ++v)
          ph[v] += silu_f(acc[m][nt][v] + cb2v[nt]) * w3v[nt];
#pragma unroll
      for (int v = 0; v < 8; ++v) {
        float s = ph[v];
        s += __shfl_xor(s, 1, 32); s += __shfl_xor(s, 2, 32);
        s += __shfl_xor(s, 4, 32); s += __shfl_xor(s, 8, 32);
        ph[v] = s + b3;
      }
      if (r == 0) {
        const int ii = mrow[m], ig = i0 + ii;
        const float xi0 = x[ig * 3 + 0], xi1 = x[ig * 3 + 1], xi2 = x[ig * 3 + 2];
#pragma unroll
        for (int v = 0; v < 8; ++v) {
          const int jj = v + 8 * hh, jg = j0 + jj;
          if (ig != jg) {
            const float s = ph[v] * invs[ii * 16 + jj];
            atomicAdd(&xacc[ii * 4 + 0], (x[jg * 3 + 0] - xi0) * s);
            atomicAdd(&xacc[ii * 4 + 1], (x[jg * 3 + 1] - xi1) * s);
            atomicAdd(&xacc[ii * 4 + 2], (x[jg * 3 + 2] - xi2) * s);
          }
        }
      }
    }
    if (has_edge) {
      __syncthreads();
      for (int it = 0; it < 128; ++it) {  // edge-MLP layer 1
        const int e = eb0 + it * 2;
        const float v = Pes[(e >> 4) * H + kloc] + Qes[(e & 15) * H + kloc] + d2s[e] * ew1c + eb1k;
        t1s[e * TSTR + kloc] = silu_f(v);
      }
      __syncthreads();
#pragma unroll
      for (int m = 0; m < 2; ++m)
#pragma unroll
        for (int nt = 0; nt < 8; ++nt) acc[m][nt] = (v8f){};
      for (int kk = 0; kk < H; kk += 4) {  // edge-MLP layer 2 GEMM
        const int kb = kk + 2 * hh;
        v2f a0, a1;
        { const float* p0 = &t1s[(mrow[0] * 16 + r) * TSTR + kb]; a0.x = p0[0]; a0.y = p0[1]; }
        { const float* p1 = &t1s[(mrow[1] * 16 + r) * TSTR + kb]; a1.x = p1[0]; a1.y = p1[1]; }
#pragma unroll
        for (int nt = 0; nt < 8; ++nt) {
          const float* wp = &w2es[(nt * 16 + r) * TSTR + kb];
          v2f b; b.x = wp[0]; b.y = wp[1];
          acc[0][nt] = wmma_f32(a0, b, acc[0][nt]);
          acc[1][nt] = wmma_f32(a1, b, acc[1][nt]);
        }
      }
      // m2 = silu(.+b2); att = sigmoid(m2.attw); msum += m2*att (skip diagonal)
#pragma unroll
      for (int m = 0; m < 2; ++m) {
        float ad[8];
#pragma unroll
        for (int v = 0; v < 8; ++v) ad[v] = 0.0f;
#pragma unroll
        for (int nt = 0; nt < 8; ++nt)
#pragma unroll
          for (int v = 0; v < 8; ++v) {
            const float mm = silu_f(acc[m][nt][v] + eb2v[nt]);
            acc[m][nt][v] = mm;
            ad[v] += mm * attv[nt];
          }
        const int ig = i0 + mrow[m];
#pragma unroll
        for (int v = 0; v < 8; ++v) {
          float s = ad[v];
          s += __shfl_xor(s, 1, 32); s += __shfl_xor(s, 2, 32);
          s += __shfl_xor(s, 4, 32); s += __shfl_xor(s, 8, 32);
          const int jg = j0 + v + 8 * hh;
          const float g = (ig == jg) ? 0.0f : sigmoid_f(s);
#pragma unroll
          for (int nt = 0; nt < 8; ++nt) msum[m][nt] += acc[m][nt][v] * g;
        }
      }
    }
  }
  __syncthreads();
  if (tid < 48) {
    const int ii = tid / 3, d = tid % 3;
    atomicAdd(&xdelta[(i0 + ii) * 3 + d], xacc[ii * 4 + d]);
  }
  if (has_edge) {
#pragma unroll
    for (int m = 0; m < 2; ++m)
#pragma unroll
      for (int nt = 0; nt < 8; ++nt) {
        const float v = msum[m][nt] + __shfl_xor(msum[m][nt], 16, 32);
        if (lane < 16)
          atomicAdd(&nbuf[(i0 + mrow[m]) * 256 + H + nt * 16 + lane], v);
      }
  }
}

__global__ void k_apply_x(float* __restrict__ x, const float* __restrict__ xd) {
  int i = blockIdx.x * 256 + threadIdx.x;
  if (i < NP * 3) x[i] += xd[i];
}
__global__ void k_nan(const float* __restrict__ x, int* __restrict__ flag) {
  int i = blockIdx.x * 256 + threadIdx.x;
  if (i < NP * 3 && isnan(x[i])) atomicOr(flag, 1);
}
__global__ void k_out(const float* __restrict__ x, const int* __restrict__ flag,
                      float* __restrict__ out) {
  int i = blockIdx.x * 256 + threadIdx.x;
  if (i < NP * 3) out[i] = (*flag) ? 0.0f : x[i];
}

// ---------------- host --------------------------------------------------------------
extern "C" void kernel_launch(void* const* d_in, const int* in_sizes, int n_in,
                              void* d_out, int out_size, void* d_ws, size_t ws_size,
                              hipStream_t stream) {
  const float* x_inp = (const float*)d_in[0];
  const float* emb_w = (const float*)d_in[1];
  const float* emb_b = (const float*)d_in[2];
  const float* cw1 = (const float*)d_in[3];
  const float* cb1 = (const float*)d_in[4];
  const float* cw2 = (const float*)d_in[5];
  const float* cb2 = (const float*)d_in[6];
  const float* cw3 = (const float*)d_in[7];
  const float* cb3 = (const float*)d_in[8];
  const float* ew1 = (const float*)d_in[9];
  const float* eb1 = (const float*)d_in[10];
  const float* ew2 = (const float*)d_in[11];
  const float* eb2 = (const float*)d_in[12];
  const float* nw1 = (const float*)d_in[13];
  const float* nb1 = (const float*)d_in[14];
  const float* nw2 = (const float*)d_in[15];
  const float* nb2 = (const float*)d_in[16];
  const float* atw = (const float*)d_in[17];

  float* f = (float*)d_ws;
  float* x    = f + 0;       // 1152
  float* xd   = f + 2048;    // 1152
  int*   flag = (int*)(f + 4096);
  float* hA   = f + 8192;    // 49152
  float* hB   = f + 57344;   // 49152
  float* P    = f + 106496;  // 49152
  float* Q    = f + 155648;  // 49152
  float* Pe   = f + 204800;  // 49152
  float* Qe   = f + 253952;  // 49152
  float* nbuf = f + 303104;  // 98304 (384 x [h | msum])
  float* tmp  = f + 401408;  // 49152

  const size_t smem = (size_t)SM_FLOATS * sizeof(float);

  k_init_h<<<(NP * H) / 8, 256, 0, stream>>>(emb_w, emb_b, hA);
  k_copy_x<<<5, 256, 0, stream>>>(x_inp, x);

  float* hcur = hA;
  float* hnxt = hB;
  for (int l = 0; l < NL; ++l) {
    const int he = (l < NL - 1) ? 1 : 0;
    const int le = he ? l : 0;  // placeholder layer for unused edge params
    const float* cw1l = cw1 + (size_t)l * H * 257;
    const float* cb1l = cb1 + l * H;
    const float* cw2l = cw2 + (size_t)l * H * H;
    const float* cb2l = cb2 + l * H;
    const float* cw3l = cw3 + l * H;
    const float* cb3l = cb3 + l;
    const float* ew1l = ew1 + (size_t)le * H * 257;
    const float* eb1l = eb1 + le * H;
    const float* ew2l = ew2 + (size_t)le * H * H;
    const float* eb2l = eb2 + le * H;
    const float* atwl = atw + le * H;

    k_prep<<<NP, 256, 0, stream>>>(xd, nbuf, hcur, flag);
    k_gemm<<<dim3(NP / 16, H / 16), 32, 0, stream>>>(hcur, H, cw1l, 257, 0,  nullptr, P, H, H, 0);
    k_gemm<<<dim3(NP / 16, H / 16), 32, 0, stream>>>(hcur, H, cw1l, 257, H,  nullptr, Q, H, H, 0);
    if (he) {
      k_gemm<<<dim3(NP / 16, H / 16), 32, 0, stream>>>(hcur, H, ew1l, 257, 0, nullptr, Pe, H, H, 0);
      k_gemm<<<dim3(NP / 16, H / 16), 32, 0, stream>>>(hcur, H, ew1l, 257, H, nullptr, Qe, H, H, 0);
    }
    k_edge<<<dim3(NP / 16, 8), 256, smem, stream>>>(
        x, P, Q, Pe, Qe, cw1l, cb1l, cw2l, cb2l, cw3l, cb3l,
        ew1l, eb1l, ew2l, eb2l, atwl, xd, nbuf, he);
    k_apply_x<<<5, 256, 0, stream>>>(x, xd);
    if (he) {
      const float* nw1l = nw1 + (size_t)l * H * 256;
      const float* nb1l = nb1 + l * H;
      const float* nw2l = nw2 + (size_t)l * H * H;
      const float* nb2l = nb2 + l * H;
      k_gemm<<<dim3(NP / 16, H / 16), 32, 0, stream>>>(nbuf, 256, nw1l, 256, 0, nb1l, tmp, H, 256, 1);
      k_gemm<<<dim3(NP / 16, H / 16), 32, 0, stream>>>(tmp, H, nw2l, H, 0, nb2l, hnxt, H, H, 0);
      float* t = hcur; hcur = hnxt; hnxt = t;
    }
  }
  k_nan<<<5, 256, 0, stream>>>(x, flag);
  k_out<<<5, 256, 0, stream>>>(x, flag, (float*)d_out);
}